// CurveGrouping_66228395704518
// MI455X (gfx1250) — compile-verified
//
#include <hip/hip_runtime.h>
#include <math.h>

// Problem constants (from reference)
#define BS 8
#define CC 128    // channels
#define NP 2048   // points
#define KN 32     // neighbors per point
#define CN 128    // curves per batch
#define CL 16     // curve length
#define BN_EPS 1e-5f

#define PSTR 132  // LDS row stride (floats) for neighbor tile (pad vs 64 banks)
#define QSTR 132  // LDS row stride (floats) for query tile

typedef float v2f __attribute__((ext_vector_type(2)));
typedef float v8f __attribute__((ext_vector_type(8)));

__device__ __forceinline__ float wred(float v) {
#pragma unroll
  for (int off = 16; off > 0; off >>= 1) v += __shfl_xor(v, off, 32);
  return v;
}

__device__ __forceinline__ float dot4(const float4& a, const float4& b) {
  return a.x * b.x + a.y * b.y + a.z * b.z + a.w * b.w;
}

// ---------------------------------------------------------------------------
// Kernel 1: per-point attention + curve-independent score components.
//   att = sigmoid(x[:,pt] . att_w)
//   s1g = att * (x[:,pt] . agent_w[:C])      (== pick . agent_w[:C] on weighted x)
//   sqn = att^2 * ||x[:,pt]||^2              (== ||pick||^2 on weighted x)
// ---------------------------------------------------------------------------
__global__ __launch_bounds__(256) void k_point_stats(
    const float* __restrict__ x, const float* __restrict__ att_w,
    const float* __restrict__ agent_w, float* __restrict__ attv,
    float* __restrict__ s1g, float* __restrict__ sqn) {
  int t = blockIdx.x * 256 + threadIdx.x;
  int b = t / NP, pt = t % NP;
  const float* xb = x + (size_t)b * CC * NP + pt;
  float a = 0.f, g = 0.f, q = 0.f;
#pragma unroll 4
  for (int ch = 0; ch < CC; ++ch) {
    float v = xb[(size_t)ch * NP];
    a += v * att_w[ch];
    g += v * agent_w[ch];
    q += v * v;
  }
  float att = 1.f / (1.f + expf(-a));
  attv[t] = att;
  s1g[t]  = g * att;
  sqn[t]  = q * att * att;
}

// ---------------------------------------------------------------------------
// Kernel 2: xw[b][pt][ch] = x[b][ch][pt] * att[b][pt]  (LDS tile transpose,
// coalesced reads and writes; point-major layout makes every later gather a
// single contiguous 512B row per point)
// ---------------------------------------------------------------------------
__global__ __launch_bounds__(256) void k_reweight_transpose(
    const float* __restrict__ x, const float* __restrict__ attv,
    float* __restrict__ xw) {
  __shared__ float tile[64 * 129];
  int b = blockIdx.x >> 5;            // 32 tiles of 64 points per batch
  int pt0 = (blockIdx.x & 31) * 64;
  int tid = threadIdx.x;
  for (int i = tid; i < 64 * CC; i += 256) {
    int ch = i >> 6, p = i & 63;
    float v = x[((size_t)b * CC + ch) * NP + pt0 + p] * attv[b * NP + pt0 + p];
    tile[p * 129 + ch] = v;
  }
  __syncthreads();
  for (int i = tid; i < 64 * CC; i += 256) {
    int p = i >> 7, ch = i & 127;
    xw[((size_t)b * NP + pt0 + p) * CC + ch] = tile[p * 129 + ch];
  }
}

// ---------------------------------------------------------------------------
// Kernel 3: sorted top-128 start points per batch (bitonic sort of 2048 keys;
// key = descending value, ascending index on ties, matching lax.top_k order)
// ---------------------------------------------------------------------------
__global__ __launch_bounds__(1024) void k_topk(const float* __restrict__ attv,
                                               int* __restrict__ start) {
  __shared__ unsigned long long keys[NP];
  int b = blockIdx.x, tid = threadIdx.x;
  for (int i = tid; i < NP; i += 1024) {
    unsigned u = __float_as_uint(attv[b * NP + i]);
    unsigned m = (u & 0x80000000u) ? ~u : (u | 0x80000000u);  // monotonic map
    keys[i] = ((unsigned long long)(~m) << 32) | (unsigned)i; // asc sort => desc value
  }
  __syncthreads();
  for (int k = 2; k <= NP; k <<= 1) {
    for (int j = k >> 1; j > 0; j >>= 1) {
      for (int base = 0; base < NP; base += 1024) {
        int i = base + tid;
        int ixj = i ^ j;
        if (ixj > i) {
          bool up = ((i & k) == 0);
          unsigned long long a = keys[i], c = keys[ixj];
          if ((a > c) == up) { keys[i] = c; keys[ixj] = a; }
        }
      }
      __syncthreads();
    }
  }
  if (tid < CN) start[b * CN + tid] = (int)(keys[tid] & 0xFFFFFFFFu);
}

// ---------------------------------------------------------------------------
// Kernel 4: curve walk. One wave32 per curve (grid = BS*CN, block = 32).
// Lane L owns channels [4L, 4L+4). Per step: momentum blend, async-DMA gather
// of 32 neighbor rows into LDS (GLOBAL_LOAD_ASYNC_TO_LDS_B128, ASYNCcnt),
// S = P(32x128) x Q(128x2) via V_WMMA_F32_16X16X4_F32 (exact fp32), crossover
// cosine from the two WMMA columns, argmax, advance.
// ---------------------------------------------------------------------------
__global__ __launch_bounds__(32) void k_curves(
    const float* __restrict__ xw, const int* __restrict__ idx,
    const float* __restrict__ s1g, const float* __restrict__ sqn,
    const int* __restrict__ start, const float* __restrict__ agent_w,
    const float* __restrict__ agent_bn, const float* __restrict__ mom_w,
    const float* __restrict__ mom_bn, float* __restrict__ out) {
  __shared__ float pick[KN * PSTR];  // 32 neighbor feature rows
  __shared__ float qm[16 * QSTR];    // query matrix rows (row0=cur_f, row1=new_pre)
  __shared__ float dots[2 * KN];     // WMMA result columns 0,1
  __shared__ int nbrs[KN];

  const int w = blockIdx.x;
  const int b = w >> 7, cv = w & (CN - 1);
  const int lane = threadIdx.x;
  const int half = lane >> 4, r = lane & 15;
  const int ch0 = lane * 4;

  // zero query rows 2..15 (only rows 0,1 carry data)
#pragma unroll
  for (int row = 2; row < 16; ++row) {
    float4 z; z.x = z.y = z.z = z.w = 0.f;
    *(float4*)&qm[row * QSTR + ch0] = z;
  }

  // BN constants (agent_bn: [g,b,m,v]; mom_bn (4,2): [g0,g1,b0,b1,m0,m1,v0,v1])
  const float ag_scale = agent_bn[0] * rsqrtf(agent_bn[3] + BN_EPS);
  const float ag_shift = agent_bn[1] - agent_bn[2] * ag_scale;
  const float ms0 = mom_bn[0] * rsqrtf(mom_bn[6] + BN_EPS);
  const float ms1 = mom_bn[1] * rsqrtf(mom_bn[7] + BN_EPS);
  const float msh0 = mom_bn[2] - mom_bn[4] * ms0;
  const float msh1 = mom_bn[3] - mom_bn[5] * ms1;

  // per-lane weight slices
  const float4 aw2  = *(const float4*)&agent_w[CC + ch0];
  const float4 mwc0 = *(const float4*)&mom_w[ch0];
  const float4 mwp0 = *(const float4*)&mom_w[CC + ch0];
  const float4 mwc1 = *(const float4*)&mom_w[2 * CC + ch0];
  const float4 mwp1 = *(const float4*)&mom_w[3 * CC + ch0];

  // per-lane, loop-invariant bases for the async gather
  const float* gbase = xw + (size_t)b * NP * CC + ch0;     // + p*CC per row
  const unsigned lds_pick0 = (unsigned)(uintptr_t)&pick[ch0];

  // ---- step 0: pre == cur == start features; no momentum / crossover ----
  int cur_idx = start[b * CN + cv];
  float4 cf = *(const float4*)&xw[((size_t)b * NP + cur_idx) * CC + ch0];
  float4 pf = cf;
  float pa = wred(dot4(cf, aw2));  // start_f . agent_w[C:]

  int nb = idx[((size_t)b * NP + cur_idx) * KN + lane];
  float sc = (s1g[b * NP + nb] + pa) * ag_scale + ag_shift;
  int wi = lane; float wv = sc;
#pragma unroll
  for (int off = 16; off > 0; off >>= 1) {
    float ov = __shfl_xor(wv, off, 32);
    int   oi = __shfl_xor(wi, off, 32);
    if (ov > wv || (ov == wv && oi < wi)) { wv = ov; wi = oi; }
  }
  int win_pt = __shfl(nb, wi, 32);
  cur_idx = win_pt;
  cf = *(const float4*)&xw[((size_t)b * NP + win_pt) * CC + ch0];

  const size_t obase = (((size_t)b * CC + ch0) * CN + cv) * CL;  // out[b][ch][cv][l]
  out[obase]               = cf.x;
  out[obase + CN * CL]     = cf.y;
  out[obase + 2 * CN * CL] = cf.z;
  out[obase + 3 * CN * CL] = cf.w;

  // ---- steps 1..15 ----
  for (int l = 1; l < CL; ++l) {
    // dynamic momentum blend
    float l0 = wred(dot4(cf, mwc0) + dot4(pf, mwp0));
    float l1 = wred(dot4(cf, mwc1) + dot4(pf, mwp1));
    float z0 = l0 * ms0 + msh0;
    float z1 = l1 * ms1 + msh1;
    float a0 = 1.f / (1.f + expf(z1 - z0));
    float a1 = 1.f - a0;
    float4 np4;
    np4.x = a0 * cf.x + a1 * pf.x;  np4.y = a0 * cf.y + a1 * pf.y;
    np4.z = a0 * cf.z + a1 * pf.z;  np4.w = a0 * cf.w + a1 * pf.w;

    // neighbor list first so the async DMA can start as early as possible
    nb = idx[((size_t)b * NP + cur_idx) * KN + lane];
    nbrs[lane] = nb;
    __syncthreads();

    // async gather: 32 neighbor rows DMA'd straight into LDS (no VGPR trip).
    // Each lane moves its 16B column chunk of row v; ASYNCcnt-tracked.
#pragma unroll 4
    for (int v = 0; v < KN; ++v) {
      int p = nbrs[v];
      unsigned long long ga = (unsigned long long)(uintptr_t)(gbase + (size_t)p * CC);
      unsigned la = lds_pick0 + (unsigned)(v * (PSTR * 4));
      asm volatile("global_load_async_to_lds_b128 %0, %1, off"
                   :: "v"(la), "v"(ga) : "memory");
    }

    // overlap with DMA: per-curve scalars + query staging
    float cc  = wred(dot4(cf, cf));
    float cp  = wred(dot4(cf, np4));
    float pp  = wred(dot4(np4, np4));
    float pa2 = wred(dot4(np4, aw2));
    *(float4*)&qm[ch0]        = cf;   // row 0 = cur_f
    *(float4*)&qm[QSTR + ch0] = np4;  // row 1 = new_pre

    // DMA must have landed in LDS before WMMA operand reads
    asm volatile("s_wait_asynccnt 0" ::: "memory");
    __syncthreads();

    // S = P(32x128) x Q(128x16, cols 0..1 live) via fp32 WMMA, K chunks of 4
    v8f acc0 = {0.f, 0.f, 0.f, 0.f, 0.f, 0.f, 0.f, 0.f};
    v8f acc1 = {0.f, 0.f, 0.f, 0.f, 0.f, 0.f, 0.f, 0.f};
    const int colsel = 2 * half;  // A/B K-halves live in lane halves
#pragma unroll
    for (int j = 0; j < 32; ++j) {
      v2f a0v = *(const v2f*)&pick[r * PSTR + 4 * j + colsel];
      v2f a1v = *(const v2f*)&pick[(16 + r) * PSTR + 4 * j + colsel];
      v2f bv  = *(const v2f*)&qm[r * QSTR + 4 * j + colsel];
      acc0 = __builtin_amdgcn_wmma_f32_16x16x4_f32(false, a0v, false, bv,
                                                   (short)0, acc0, false, false);
      acc1 = __builtin_amdgcn_wmma_f32_16x16x4_f32(false, a1v, false, bv,
                                                   (short)0, acc1, false, false);
    }
    // D layout: VGPR rr, lane = 16*half2 + col -> row = rr + 8*half2
    if (r < 2) {
#pragma unroll
      for (int rr = 0; rr < 8; ++rr) {
        dots[r * KN + 8 * half + rr]      = acc0[rr];
        dots[r * KN + 16 + 8 * half + rr] = acc1[rr];
      }
    }
    __syncthreads();

    float dc = dots[lane];       // pick_k . cur_f
    float dp = dots[KN + lane];  // pick_k . new_pre
    float s1v = s1g[b * NP + nb];
    float qn  = sqn[b * NP + nb];
    float sc2 = (s1v + pa2) * ag_scale + ag_shift;
    // crossover suppression: cur_dir=(cf-npre), nbr_dir=(pick-cf)
    float dotv  = (dc - dp) - (cc - cp);
    float normc = sqrtf(fmaxf(cc - 2.f * cp + pp, 0.f));
    float normn = sqrtf(fmaxf(qn - 2.f * dc + cc, 0.f));
    float dd = fminf(fmaxf(1.f + dotv / fmaxf(normc * normn, 1e-8f), 0.f), 1.f);
    sc2 *= dd;

    int wi2 = lane; float wv2 = sc2;
#pragma unroll
    for (int off = 16; off > 0; off >>= 1) {
      float ov = __shfl_xor(wv2, off, 32);
      int   oi = __shfl_xor(wi2, off, 32);
      if (ov > wv2 || (ov == wv2 && oi < wi2)) { wv2 = ov; wi2 = oi; }
    }
    int wpt = nbrs[wi2];
    float4 nc = *(const float4*)&pick[wi2 * PSTR + ch0];

    pf = np4; cf = nc; cur_idx = wpt;
    out[obase + l]               = cf.x;
    out[obase + CN * CL + l]     = cf.y;
    out[obase + 2 * CN * CL + l] = cf.z;
    out[obase + 3 * CN * CL + l] = cf.w;
    __syncthreads();  // drain DS reads before next iteration's DMA overwrites
  }
}

// ---------------------------------------------------------------------------
extern "C" void kernel_launch(void* const* d_in, const int* in_sizes, int n_in,
                              void* d_out, int out_size, void* d_ws, size_t ws_size,
                              hipStream_t stream) {
  const float* x        = (const float*)d_in[0];
  // d_in[1] = xyz (unused by the reference computation)
  const int*   idx      = (const int*)d_in[2];
  const float* att_w    = (const float*)d_in[3];
  const float* agent_w  = (const float*)d_in[4];
  const float* agent_bn = (const float*)d_in[5];
  const float* mom_w    = (const float*)d_in[6];
  const float* mom_bn   = (const float*)d_in[7];
  float* out = (float*)d_out;

  float* xw   = (float*)d_ws;                      // BS*NP*CC  (8 MB)
  float* attv = xw + (size_t)BS * NP * CC;         // BS*NP
  float* s1g  = attv + (size_t)BS * NP;            // BS*NP
  float* sqn  = s1g + (size_t)BS * NP;             // BS*NP
  int*   st   = (int*)(sqn + (size_t)BS * NP);     // BS*CN

  k_point_stats<<<BS * NP / 256, 256, 0, stream>>>(x, att_w, agent_w, attv, s1g, sqn);
  k_reweight_transpose<<<BS * (NP / 64), 256, 0, stream>>>(x, attv, xw);
  k_topk<<<BS, 1024, 0, stream>>>(attv, st);
  k_curves<<<BS * CN, 32, 0, stream>>>(xw, idx, s1g, sqn, st, agent_w, agent_bn,
                                       mom_w, mom_bn, out);
}